// FeatureFusionLayer_91250875171161
// MI455X (gfx1250) — compile-verified
//

#include <hip/hip_runtime.h>

// FeatureFusionLayer for MI455X (gfx1250, wave32).
// N=262144 nodes, C=256 channels, P=4096 patches (constants match reference).
// Strategy: all GEMMs via V_WMMA_F32_16X16X4_F32 (fp32-exact path).
// Big fuse GEMM is reduced from K=512 to K=256 by precomputing
// dgW = dg @ Wp.T + fuse_b at patch granularity, and LN(original) is fused
// into the big kernel via an LDS tile so `original` is read exactly once.

#define C 256
#define NNODES 262144
#define NPATCH 4096

typedef float v2f __attribute__((ext_vector_type(2)));
typedef float v8f __attribute__((ext_vector_type(8)));

__device__ __forceinline__ float wave_sum32(float v) {
#pragma unroll
    for (int m = 16; m >= 1; m >>= 1) v += __shfl_xor(v, m, 32);
    return v;
}

// ---------------------------------------------------------------------------
// K1: segment mean over sorted `batch` -> dg_global[P][C]
// One block per patch; binary search gives the contiguous row range.
// ---------------------------------------------------------------------------
__global__ void seg_mean_kernel(const float* __restrict__ dgf,
                                const int* __restrict__ batch,
                                float* __restrict__ dg_global) {
    int p = blockIdx.x;
    int lo = 0, hi = NNODES;
    while (lo < hi) { int mid = (lo + hi) >> 1; if (batch[mid] < p) lo = mid + 1; else hi = mid; }
    int start = lo;
    hi = NNODES;
    while (lo < hi) { int mid = (lo + hi) >> 1; if (batch[mid] < p + 1) lo = mid + 1; else hi = mid; }
    int end = lo;
    int c = threadIdx.x;
    float acc = 0.f;
    for (int r = start; r < end; ++r) acc += dgf[(size_t)r * C + c];
    float cnt = (float)(end - start);
    dg_global[(size_t)p * C + c] = (cnt > 0.f) ? (acc / cnt) : 0.f;
}

// ---------------------------------------------------------------------------
// K2/K5: row LayerNorm over C=256, one wave per row (8 rows / 256-thr block).
// Safe in-place (each element read+written by the same lane).
// ---------------------------------------------------------------------------
__global__ void ln_rows_kernel(const float* __restrict__ in, float* __restrict__ out,
                               const float* __restrict__ w, const float* __restrict__ b) {
    int wave = threadIdx.x >> 5, lane = threadIdx.x & 31;
    int row = blockIdx.x * 8 + wave;
    const float* src = in + (size_t)row * C;
    float v[8];
#pragma unroll
    for (int i = 0; i < 8; ++i) v[i] = src[lane + 32 * i];
    float s = 0.f;
#pragma unroll
    for (int i = 0; i < 8; ++i) s += v[i];
    float mu = wave_sum32(s) * (1.f / C);
    float q = 0.f;
#pragma unroll
    for (int i = 0; i < 8; ++i) { float d = v[i] - mu; q += d * d; }
    float inv = rsqrtf(wave_sum32(q) * (1.f / C) + 1e-5f);
    float* dst = out + (size_t)row * C;
#pragma unroll
    for (int i = 0; i < 8; ++i) {
        int c = lane + 32 * i;
        dst[c] = (v[i] - mu) * inv * w[c] + b[c];
    }
}

// ---------------------------------------------------------------------------
// K3..K8: small WMMA GEMM: out[M x C] = A[M x C] @ W(.,wOff:wOff+C).T + bias
//         (+ optional addend, optional relu).  One wave per 16x16 tile.
// Grid: (C/16, M/16), block 32.  All operands are L2-resident (<= 4 MB).
// ---------------------------------------------------------------------------
__global__ void gemm_wmma_kernel(const float* __restrict__ A,
                                 const float* __restrict__ W, int ldw, int wOff,
                                 const float* __restrict__ bias,
                                 const float* __restrict__ addend, int doRelu,
                                 float* __restrict__ out) {
    int lane = threadIdx.x;
    int n0 = blockIdx.x * 16, m0 = blockIdx.y * 16;
    int g = lane >> 4, l = lane & 15;
    const float* arow = A + (size_t)(m0 + l) * C;
    const float* wrow = W + (size_t)(n0 + l) * ldw + wOff;
    v8f acc = {0.f, 0.f, 0.f, 0.f, 0.f, 0.f, 0.f, 0.f};
#pragma unroll 8
    for (int k = 0; k < C; k += 4) {
        v2f a = *(const v2f*)(arow + k + 2 * g);   // A[m][k+2g], A[m][k+2g+1]
        v2f bb = *(const v2f*)(wrow + k + 2 * g);  // W[n][k+2g], W[n][k+2g+1]
        acc = __builtin_amdgcn_wmma_f32_16x16x4_f32(false, a, false, bb,
                                                    (short)0, acc, false, false);
    }
    int n = n0 + l;
    float bn = bias ? bias[n] : 0.f;
#pragma unroll
    for (int r = 0; r < 8; ++r) {
        int m = m0 + r + 8 * g;
        float val = acc[r] + bn;
        if (addend) val += addend[(size_t)m * C + n];
        if (doRelu) val = fmaxf(val, 0.f);
        out[(size_t)m * C + n] = val;
    }
}

// ---------------------------------------------------------------------------
// K9: fused node kernel:  out = relu( LN(orig) @ Wx.T + dgW[batch] ) + LN(orig)
// Block = 256 threads (8 waves), 64-row tile. LN result lives in LDS only.
// Wave w: row-group (w>>1)*16, col half (w&1)*128 -> 8 WMMA accum tiles.
// ---------------------------------------------------------------------------
#define MB 64
#define KCHUNK 32
#define XSTR 260   // pad: conflict-free b64 A-fragment LDS reads
#define WSTR 36    // pad: conflict-free b64 B-fragment LDS reads

__launch_bounds__(256, 1)
__global__ void fuse_kernel(const float* __restrict__ orig,
                            const float* __restrict__ nw, const float* __restrict__ nb,
                            const float* __restrict__ fuse_w,  // (C, 2C) row-major
                            const int* __restrict__ batch,
                            const float* __restrict__ dgW,     // P x C, fuse_b folded in
                            float* __restrict__ out) {
    __shared__ float xs[MB * XSTR];
    __shared__ float wsh[C * WSTR];
    __shared__ int bl[MB];

    int tid = threadIdx.x, wave = tid >> 5, lane = tid & 31;
    int rowBase = blockIdx.x * MB;

    // ---- Phase 1: LayerNorm 64 rows into LDS (wave-per-row, 8 rows/wave) ---
#pragma unroll
    for (int rr = 0; rr < 8; ++rr) {
        int row = wave * 8 + rr;
        const float* src = orig + (size_t)(rowBase + row) * C;
        float v[8];
#pragma unroll
        for (int i = 0; i < 8; ++i) v[i] = src[lane + 32 * i];
        float s = 0.f;
#pragma unroll
        for (int i = 0; i < 8; ++i) s += v[i];
        float mu = wave_sum32(s) * (1.f / C);
        float q = 0.f;
#pragma unroll
        for (int i = 0; i < 8; ++i) { float d = v[i] - mu; q += d * d; }
        float inv = rsqrtf(wave_sum32(q) * (1.f / C) + 1e-5f);
#pragma unroll
        for (int i = 0; i < 8; ++i) {
            int c = lane + 32 * i;
            xs[row * XSTR + c] = (v[i] - mu) * inv * nw[c] + nb[c];
        }
    }
    if (tid < MB) bl[tid] = batch[rowBase + tid];

    // ---- Phase 2: GEMM  x(64x256) @ Wx.T(256x256) via WMMA f32 ------------
    int rg = wave >> 1;              // row group 0..3 (16 rows each)
    int n0w = (wave & 1) * 128;      // column half
    int g = lane >> 4, l = lane & 15;
    v8f acc[8];
#pragma unroll
    for (int j = 0; j < 8; ++j) acc[j] = (v8f){0.f,0.f,0.f,0.f,0.f,0.f,0.f,0.f};

    for (int kc = 0; kc < C; kc += KCHUNK) {
        __syncthreads();  // prior chunk consumed (and Phase-1 xs visible)
        // stage W chunk (256 rows x 32 k) coalesced: float4 per thread x8
#pragma unroll
        for (int i = 0; i < 8; ++i) {
            int idx = tid + 256 * i;
            int n = idx >> 3, k4 = idx & 7;
            float4 wv = *(const float4*)(fuse_w + (size_t)n * (2 * C) + kc + k4 * 4);
            *(float4*)&wsh[n * WSTR + k4 * 4] = wv;
        }
        __syncthreads();
        const float* xrow = &xs[(rg * 16 + l) * XSTR];
#pragma unroll
        for (int kk = 0; kk < KCHUNK; kk += 4) {
            v2f a = *(const v2f*)(xrow + kc + kk + 2 * g);
#pragma unroll
            for (int j = 0; j < 8; ++j) {
                v2f bb = *(const v2f*)&wsh[(n0w + j * 16 + l) * WSTR + kk + 2 * g];
                acc[j] = __builtin_amdgcn_wmma_f32_16x16x4_f32(false, a, false, bb,
                                                               (short)0, acc[j], false, false);
            }
        }
    }

    // ---- Phase 3: epilogue  relu(acc + dgW[batch]) + x  --------------------
#pragma unroll
    for (int j = 0; j < 8; ++j) {
        int n = n0w + j * 16 + l;
#pragma unroll
        for (int r = 0; r < 8; ++r) {
            int mloc = rg * 16 + r + 8 * g;
            float val = acc[j][r] + dgW[(size_t)bl[mloc] * C + n];
            val = fmaxf(val, 0.f) + xs[mloc * XSTR + n];
            out[(size_t)(rowBase + mloc) * C + n] = val;
        }
    }
}

// ---------------------------------------------------------------------------
extern "C" void kernel_launch(void* const* d_in, const int* in_sizes, int n_in,
                              void* d_out, int out_size, void* d_ws, size_t ws_size,
                              hipStream_t stream) {
    const float* orig       = (const float*)d_in[0];
    const float* dgf        = (const float*)d_in[1];
    const float* node_w     = (const float*)d_in[2];
    const float* node_b     = (const float*)d_in[3];
    const float* dg_w       = (const float*)d_in[4];
    const float* dg_b       = (const float*)d_in[5];
    const float* in_proj_w  = (const float*)d_in[6];
    const float* in_proj_b  = (const float*)d_in[7];
    const float* out_proj_w = (const float*)d_in[8];
    const float* out_proj_b = (const float*)d_in[9];
    const float* ffn_w      = (const float*)d_in[10];
    const float* ffn_b      = (const float*)d_in[11];
    const float* lin1_w     = (const float*)d_in[12];
    const float* lin1_b     = (const float*)d_in[13];
    const float* lin2_w     = (const float*)d_in[14];
    const float* lin2_b     = (const float*)d_in[15];
    const float* fuse_w     = (const float*)d_in[16];
    const float* fuse_b     = (const float*)d_in[17];
    const int*   batch      = (const int*)d_in[18];
    float* out = (float*)d_out;

    // workspace: 4 buffers of P*C fp32 = 16 MB
    float* b0 = (float*)d_ws;
    float* b1 = b0 + (size_t)NPATCH * C;
    float* b2 = b1 + (size_t)NPATCH * C;
    float* b3 = b2 + (size_t)NPATCH * C;

    // Patch pipeline (all tiny / L2-resident)
    seg_mean_kernel<<<NPATCH, C, 0, stream>>>(dgf, batch, b0);            // b0 = dg_global
    ln_rows_kernel<<<NPATCH / 8, 256, 0, stream>>>(b0, b0, dg_w, dg_b);   // b0 = g (in-place)
    dim3 gs(C / 16, NPATCH / 16);
    // v = g @ in_proj_w[2C:3C].T + in_proj_b[2C:3C]
    gemm_wmma_kernel<<<gs, 32, 0, stream>>>(b0, in_proj_w + 2 * C * C, C, 0,
                                            in_proj_b + 2 * C, nullptr, 0, b1);
    // attn = v @ out_proj_w.T + out_proj_b     (kept as residual in b2)
    gemm_wmma_kernel<<<gs, 32, 0, stream>>>(b1, out_proj_w, C, 0,
                                            out_proj_b, nullptr, 0, b2);
    ln_rows_kernel<<<NPATCH / 8, 256, 0, stream>>>(b2, b1, ffn_w, ffn_b); // b1 = y
    // h = relu(y @ lin1_w.T + lin1_b)
    gemm_wmma_kernel<<<gs, 32, 0, stream>>>(b1, lin1_w, C, 0,
                                            lin1_b, nullptr, 1, b3);
    // dg = h @ lin2_w.T + lin2_b + attn
    gemm_wmma_kernel<<<gs, 32, 0, stream>>>(b3, lin2_w, C, 0,
                                            lin2_b, b2, 0, b1);
    // dgW = dg @ Wp.T + fuse_b   (Wp = fuse_w[:, C:2C], row stride 2C)
    gemm_wmma_kernel<<<gs, 32, 0, stream>>>(b1, fuse_w, 2 * C, C,
                                            fuse_b, nullptr, 0, b0);

    // Big fused node kernel: LN + GEMM(K=256) + gather + relu + residual
    fuse_kernel<<<NNODES / MB, 256, 0, stream>>>(orig, node_w, node_b,
                                                 fuse_w, batch, b0, out);
}